// VQCParamVQC_86096914415919
// MI455X (gfx1250) — compile-verified
//
#include <hip/hip_runtime.h>
#include <hip/hip_bf16.h>
#include <math.h>

#define N_WIRES  12
#define N_QLAY   2
#define QDIM     4096
#define BSZ      2048
#define XCOLS    1024
#define TTN      72
#define WPAD     80

typedef float v2f __attribute__((ext_vector_type(2)));
typedef float v8f __attribute__((ext_vector_type(8)));

// ---------------------------------------------------------------------------
// Kernel 1: pre-contract core0 (1,32,8,16) and core1 (16,32,9,1) over q into
// W[(i*32+j)][(p*9+r)], stored 1024 x 80 (cols 72..79 zero padding).
// ---------------------------------------------------------------------------
__global__ void build_w_kernel(const float* __restrict__ core0,
                               const float* __restrict__ core1,
                               float* __restrict__ W) {
    int t = blockIdx.x * blockDim.x + threadIdx.x;
    if (t >= 1024 * WPAD) return;
    int ij = t / WPAD, pr = t % WPAD;
    float acc = 0.f;
    if (pr < TTN) {
        int p = pr / 9, r = pr % 9;
        int i = ij >> 5, j = ij & 31;
        #pragma unroll
        for (int q = 0; q < 16; ++q)
            acc += core0[i * 128 + p * 16 + q] * core1[(q * 32 + j) * 9 + r];
    }
    W[t] = acc;
}

// ---------------------------------------------------------------------------
// Kernel 2: tt_out[b][n] = sum_k x[b][k] * W[k][n] + bias[n]
// f32 WMMA 16x16x4. One wave per 16x16 output tile.
// M=2048 -> 128 tiles, N=80(padded) -> 5 tiles, 640 waves = 80 blocks x 8 waves
// ---------------------------------------------------------------------------
__global__ void tt_gemm_kernel(const float* __restrict__ x,
                               const float* __restrict__ W,
                               const float* __restrict__ bias,
                               float* __restrict__ tt_out) {
    int wid  = (blockIdx.x * blockDim.x + threadIdx.x) >> 5;
    int lane = threadIdx.x & 31;
    int mt = wid / 5, nt = wid % 5;
    int m0 = mt * 16, n0 = nt * 16;
    int half = lane >> 4;       // 0: K pair {0,1}, 1: K pair {2,3}
    int l    = lane & 15;

    // A: row (m0+l), columns k0 + 2*half + {0,1}  (8B aligned -> vector load)
    const float* arow = x + (size_t)(m0 + l) * XCOLS + 2 * half;
    // B: col (n0+l), rows k0 + 2*half + {0,1}
    const float* bcol = W + (2 * half) * WPAD + n0 + l;

    v8f c = {};
    for (int k0 = 0; k0 < XCOLS; k0 += 4) {
        v2f a = *(const v2f*)(arow + k0);
        v2f b;
        b.x = bcol[k0 * WPAD];
        b.y = bcol[k0 * WPAD + WPAD];
        c = __builtin_amdgcn_wmma_f32_16x16x4_f32(
                /*neg_a=*/false, a, /*neg_b=*/false, b,
                /*c_mod=*/(short)0, c, /*reuse_a=*/false, /*reuse_b=*/false);
    }

    int col = n0 + l;
    if (col < TTN) {
        float bv = bias[col];
        #pragma unroll
        for (int r = 0; r < 8; ++r) {
            int row = m0 + r + 8 * half;          // C/D layout: VGPR r -> M=r (+8 upper half)
            tt_out[(size_t)row * TTN + col] = c[r] + bv;
        }
    }
}

// ---------------------------------------------------------------------------
// Kernel 3: 12-qubit statevector simulation, one block (256 thr) per sample.
// State (4096 complex) lives in LDS. Wire w <-> bit (11-w).
// ---------------------------------------------------------------------------
__global__ void __launch_bounds__(256)
vqc_sim_kernel(const float* __restrict__ x,
               const float* __restrict__ tt_out,
               const float* __restrict__ gang,
               float* __restrict__ out) {
    __shared__ float sre[QDIM];
    __shared__ float sim_[QDIM];
    __shared__ float uM[24][8];
    __shared__ float ampc[N_WIRES], amps[N_WIRES];

    int b   = blockIdx.x;
    int tid = threadIdx.x;

    // Amplitude-encoding factors
    if (tid < N_WIRES) {
        float e = x[(size_t)b * XCOLS + tid] * 0.5f;
        ampc[tid] = cosf(e);
        amps[tid] = sinf(e);
    }
    // Gate unitaries U = Rz(az)*Ry(ay)*Rx(ax), angles = global + tt_out row
    if (tid < 24) {
        int g = tid;                                   // g = k*12 + w
        const float* tb = tt_out + (size_t)b * TTN + g * 3;
        const float* gb = gang + g * 3;
        float ax = (gb[0] + tb[0]) * 0.5f;
        float ay = (gb[1] + tb[1]) * 0.5f;
        float az = (gb[2] + tb[2]) * 0.5f;
        float cx = cosf(ax), sx = sinf(ax);
        float cy = cosf(ay), sy = sinf(ay);
        float cz = cosf(az), sz = sinf(az);
        // Ry*Rx
        float m00r =  cy * cx, m00i =  sy * sx;
        float m01r = -sy * cx, m01i = -cy * sx;
        float m10r =  sy * cx, m10i = -cy * sx;
        float m11r =  cy * cx, m11i = -sy * sx;
        // row0 *= (cz - i sz) ; row1 *= (cz + i sz)
        uM[g][0] = cz * m00r + sz * m00i;  uM[g][1] = cz * m00i - sz * m00r;
        uM[g][2] = cz * m01r + sz * m01i;  uM[g][3] = cz * m01i - sz * m01r;
        uM[g][4] = cz * m10r - sz * m10i;  uM[g][5] = cz * m10i + sz * m10r;
        uM[g][6] = cz * m11r - sz * m11i;  uM[g][7] = cz * m11i + sz * m11r;
    }
    __syncthreads();

    // Product state: state[idx] = prod_w amp[w][bit_{11-w}(idx)]
    for (int t = 0; t < 16; ++t) {
        int idx = tid + t * 256;
        float v = 1.f;
        #pragma unroll
        for (int w = 0; w < N_WIRES; ++w)
            v *= ((idx >> (11 - w)) & 1) ? amps[w] : ampc[w];
        sre[idx]  = v;
        sim_[idx] = 0.f;
    }
    __syncthreads();

    for (int k = 0; k < N_QLAY; ++k) {
        // single-qubit gates
        for (int w = 0; w < N_WIRES; ++w) {
            int g = k * 12 + w;
            float u00r = uM[g][0], u00i = uM[g][1];
            float u01r = uM[g][2], u01i = uM[g][3];
            float u10r = uM[g][4], u10i = uM[g][5];
            float u11r = uM[g][6], u11i = uM[g][7];
            int mask = 1 << (11 - w);
            #pragma unroll
            for (int t = 0; t < 8; ++t) {
                int p   = tid + t * 256;                 // pair index 0..2047
                int low = p & (mask - 1);
                int i0  = ((p - low) << 1) | low;
                int i1  = i0 | mask;
                float r0 = sre[i0], q0 = sim_[i0];
                float r1 = sre[i1], q1 = sim_[i1];
                sre[i0]  = u00r * r0 - u00i * q0 + u01r * r1 - u01i * q1;
                sim_[i0] = u00r * q0 + u00i * r0 + u01r * q1 + u01i * r1;
                sre[i1]  = u10r * r0 - u10i * q0 + u11r * r1 - u11i * q1;
                sim_[i1] = u10r * q0 + u10i * r0 + u11r * q1 + u11i * r1;
            }
            __syncthreads();
        }
        // CNOT chain (0->1, ..., 10->11, 11->0): swap pairs along target bit
        for (int i = 0; i < 12; ++i) {
            int cw = (i < 11) ? i : 11;
            int tw = (i < 11) ? i + 1 : 0;
            int mc = 1 << (11 - cw), mt = 1 << (11 - tw);
            #pragma unroll
            for (int t = 0; t < 16; ++t) {
                int idx = tid + t * 256;
                if ((idx & mc) && !(idx & mt)) {
                    int j = idx | mt;
                    float a0 = sre[idx], a1 = sim_[idx];
                    sre[idx] = sre[j];  sim_[idx] = sim_[j];
                    sre[j]   = a0;      sim_[j]   = a1;
                }
            }
            __syncthreads();
        }
    }

    // <Z_w> = sum_idx |amp|^2 * (1 - 2*bit_{11-w}(idx))
    float z[N_WIRES];
    #pragma unroll
    for (int w = 0; w < N_WIRES; ++w) z[w] = 0.f;
    for (int t = 0; t < 16; ++t) {
        int idx = tid + t * 256;
        float pr = sre[idx] * sre[idx] + sim_[idx] * sim_[idx];
        #pragma unroll
        for (int w = 0; w < N_WIRES; ++w)
            z[w] += (idx & (1 << (11 - w))) ? -pr : pr;
    }
    __syncthreads();   // state no longer needed; reuse sre as reduction scratch

    for (int w = 0; w < N_WIRES; ++w) {
        sre[tid] = z[w];
        __syncthreads();
        for (int s = 128; s > 0; s >>= 1) {
            if (tid < s) sre[tid] += sre[tid + s];
            __syncthreads();
        }
        if (tid == 0) out[(size_t)b * N_WIRES + w] = sre[0];
        __syncthreads();
    }
}

// ---------------------------------------------------------------------------
extern "C" void kernel_launch(void* const* d_in, const int* in_sizes, int n_in,
                              void* d_out, int out_size, void* d_ws, size_t ws_size,
                              hipStream_t stream) {
    const float* x       = (const float*)d_in[0];   // (2048,1024)
    const float* core0   = (const float*)d_in[1];   // (1,32,8,16)
    const float* core1   = (const float*)d_in[2];   // (16,32,9,1)
    const float* tt_bias = (const float*)d_in[3];   // (72,)
    const float* gang    = (const float*)d_in[4];   // (2,12,3)
    float* out = (float*)d_out;                     // (2048,12)

    float* W  = (float*)d_ws;                       // 1024*80 floats
    float* tt = W + 1024 * WPAD;                    // 2048*72 floats

    build_w_kernel<<<(1024 * WPAD + 255) / 256, 256, 0, stream>>>(core0, core1, W);
    tt_gemm_kernel<<<80, 256, 0, stream>>>(x, W, tt_bias, tt);  // 640 waves = 128x5 tiles
    vqc_sim_kernel<<<BSZ, 256, 0, stream>>>(x, tt, gang, out);
}